// GraphSAGE_61761629716974
// MI455X (gfx1250) — compile-verified
//
#include <hip/hip_runtime.h>

// ---------------------------------------------------------------------------
// GraphSAGE (mean aggregator), 3 layers, fp32 end-to-end.
// GEMMs via V_WMMA_F32_16X16X4_F32 (full-precision CDNA5 WMMA).
// Weights staged once per block into LDS; grid-stride over 16-row tiles.
// Dual independent WMMA accumulator chains (self / neigh) for matrix-pipe ILP.
// ---------------------------------------------------------------------------

typedef float v2f __attribute__((ext_vector_type(2)));
typedef float v8f __attribute__((ext_vector_type(8)));

#define DIM_IN 128          // inner dim for every layer
#define LDS_STRIDE 132      // 128 + 4 pad -> conflict-free ds_load_b64 A-fragments

// ---------------------------------------------------------------- utilities
__global__ void zero_f32(float* __restrict__ p, long long n) {
  long long i = (long long)blockIdx.x * blockDim.x + threadIdx.x;
  long long stride = (long long)gridDim.x * blockDim.x;
  for (; i < n; i += stride) p[i] = 0.0f;
}

__global__ void degree_kernel(const long long* __restrict__ dst,
                              float* __restrict__ deg, int E) {
  int e = blockIdx.x * blockDim.x + threadIdx.x;
  if (e < E) atomicAdd(&deg[dst[e]], 1.0f);
}

__global__ void inv_deg_kernel(float* __restrict__ deg, int N) {
  int v = blockIdx.x * blockDim.x + threadIdx.x;
  if (v < N) deg[v] = 1.0f / fmaxf(deg[v], 1.0f);
}

// ------------------------------------------------------------- scatter-add
// One wave per edge: 32 lanes x float4 covers the 128-float feature row.
// Gather from feat[src] (L2-resident), global_atomic_add_f32 into agg[dst].
__global__ __launch_bounds__(256) void scatter_kernel(
    const float* __restrict__ feat,
    const long long* __restrict__ src,
    const long long* __restrict__ dst,
    float* __restrict__ agg, int E) {
  long long t = (long long)blockIdx.x * blockDim.x + threadIdx.x;
  if (t >= (long long)E * 32) return;
  int e = (int)(t >> 5);
  int j = (int)(t & 31);
  long long s = src[e];
  long long d = dst[e];
  const float4 v = *(const float4*)(feat + s * DIM_IN + j * 4);
  float* a = agg + d * DIM_IN + j * 4;
  atomicAdd(a + 0, v.x);
  atomicAdd(a + 1, v.y);
  atomicAdd(a + 2, v.z);
  atomicAdd(a + 3, v.w);
}

// ------------------------------------------------------ fused SAGE layer
// out[v,:] = act( feat[v,:] @ Wself + (agg[v,:]*invd[v]) @ Wneigh + bias )
// Block: 256 threads = 8 waves. CT = DOUT/16 column tiles; RT = 8/CT row
// tiles per iteration so ALL waves do WMMAs (EXEC all-ones). Both weight
// matrices live in LDS (staged once per block); block grid-strides over
// row-tile groups, so each iteration is pure ds_load + v_wmma with two
// independent accumulator chains.
template <int DOUT, bool RELU>
__global__ __launch_bounds__(256) void sage_layer_kernel(
    const float* __restrict__ feat, const float* __restrict__ agg,
    const float* __restrict__ invd,
    const float* __restrict__ Wself, const float* __restrict__ Wneigh,
    const float* __restrict__ bias, float* __restrict__ out, int N) {
  constexpr int CT = DOUT / 16;      // column tiles (8 or 4)
  constexpr int RT = 8 / CT;         // row tiles per iteration (1 or 2)
  constexpr int WSTR = DOUT + 8;     // padded weight stride: halves hit
                                     // disjoint bank groups (2*WSTR%64==16)
  __shared__ float sWs[DIM_IN * WSTR];
  __shared__ float sWn[DIM_IN * WSTR];
  __shared__ float sFeat[RT * 16 * LDS_STRIDE];
  __shared__ float sNeigh[RT * 16 * LDS_STRIDE];

  const int tid = threadIdx.x;

  // ---- stage both weight matrices into LDS once per block (float4) ----
  for (int i = tid; i < DIM_IN * (DOUT / 4); i += 256) {
    const int r = i / (DOUT / 4);
    const int c4 = i % (DOUT / 4);
    const float4 ws = *(const float4*)(Wself + (long long)r * DOUT + c4 * 4);
    const float4 wn = *(const float4*)(Wneigh + (long long)r * DOUT + c4 * 4);
    *(float4*)&sWs[r * WSTR + c4 * 4] = ws;
    *(float4*)&sWn[r * WSTR + c4 * 4] = wn;
  }

  const int wave = tid >> 5;
  const int lane = tid & 31;
  const int colTile = wave % CT;
  const int rowTile = wave / CT;     // 0..RT-1
  const int col0 = colTile * 16;
  const int half = lane >> 4;        // 0: K even pair, 1: K odd pair
  const int m = lane & 15;           // A row / B column within tile
  const int n = col0 + m;            // global output column
  const float bn = bias[n];

  const int numIter = (N + RT * 16 - 1) / (RT * 16);

  for (int iter = blockIdx.x; iter < numIter; iter += gridDim.x) {
    const int rowBase = iter * RT * 16;

    // ---- stage RT x 16 rows of feat and mean-scaled agg into LDS ----
    for (int i = tid; i < RT * 16 * (DIM_IN / 4); i += 256) {
      const int r = i >> 5;          // row within tile group
      const int c4 = i & 31;         // float4 column
      const int row = rowBase + r;
      float4 f = make_float4(0.f, 0.f, 0.f, 0.f);
      float4 g = make_float4(0.f, 0.f, 0.f, 0.f);
      float sc = 0.f;
      if (row < N) {
        f = *(const float4*)(feat + (long long)row * DIM_IN + c4 * 4);
        g = *(const float4*)(agg + (long long)row * DIM_IN + c4 * 4);
        sc = invd[row];
      }
      float* pf = &sFeat[r * LDS_STRIDE + c4 * 4];
      pf[0] = f.x; pf[1] = f.y; pf[2] = f.z; pf[3] = f.w;
      float* pn = &sNeigh[r * LDS_STRIDE + c4 * 4];
      pn[0] = g.x * sc; pn[1] = g.y * sc; pn[2] = g.z * sc; pn[3] = g.w * sc;

      // prefetch next grid-stride iteration's rows into cache (gfx1250
      // global_prefetch_b8) while this tile computes
      const int nrow = rowBase + (int)gridDim.x * RT * 16 + r;
      if (nrow < N) {
        __builtin_prefetch(feat + (long long)nrow * DIM_IN + c4 * 4, 0, 0);
        __builtin_prefetch(agg + (long long)nrow * DIM_IN + c4 * 4, 0, 0);
      }
    }
    __syncthreads();

    // ---- 16x16 output tile per wave: 64 f32 WMMAs over K=2x128,
    //      two independent accumulator chains for WMMA-pipe ILP ----
    const float* As = &sFeat[rowTile * 16 * LDS_STRIDE];
    const float* An = &sNeigh[rowTile * 16 * LDS_STRIDE];
    v8f c0 = {};   // feat   @ Wself
    v8f c1 = {};   // h_mean @ Wneigh

#pragma unroll
    for (int k = 0; k < DIM_IN / 4; ++k) {
      const int kk = 4 * k + 2 * half;
      v2f as, bs, an, bnv;
      as.x = As[m * LDS_STRIDE + kk];
      as.y = As[m * LDS_STRIDE + kk + 1];
      bs.x = sWs[kk * WSTR + n];
      bs.y = sWs[(kk + 1) * WSTR + n];
      an.x = An[m * LDS_STRIDE + kk];
      an.y = An[m * LDS_STRIDE + kk + 1];
      bnv.x = sWn[kk * WSTR + n];
      bnv.y = sWn[(kk + 1) * WSTR + n];
      c0 = __builtin_amdgcn_wmma_f32_16x16x4_f32(false, as, false, bs,
                                                 (short)0, c0, false, false);
      c1 = __builtin_amdgcn_wmma_f32_16x16x4_f32(false, an, false, bnv,
                                                 (short)0, c1, false, false);
    }

#pragma unroll
    for (int r = 0; r < 8; ++r) {
      const int mr = r + 8 * half;   // D layout: VGPR r holds M=r (lo) / M=r+8 (hi)
      const int row = rowBase + rowTile * 16 + mr;
      if (row < N) {
        float v = c0[r] + c1[r] + bn;
        if (RELU) v = fmaxf(v, 0.f);
        out[(long long)row * DOUT + n] = v;
      }
    }
    __syncthreads();   // protect LDS tiles before next iteration's staging
  }
}

// --------------------------------------------------------------- launcher
extern "C" void kernel_launch(void* const* d_in, const int* in_sizes, int n_in,
                              void* d_out, int out_size, void* d_ws, size_t ws_size,
                              hipStream_t stream) {
  const float* features = (const float*)d_in[0];
  const long long* edge = (const long long*)d_in[1];
  const float* Ws0 = (const float*)d_in[2];
  const float* Wn0 = (const float*)d_in[3];
  const float* b0 = (const float*)d_in[4];
  const float* Ws1 = (const float*)d_in[5];
  const float* Wn1 = (const float*)d_in[6];
  const float* b1 = (const float*)d_in[7];
  const float* Ws2 = (const float*)d_in[8];
  const float* Wn2 = (const float*)d_in[9];
  const float* b2 = (const float*)d_in[10];
  float* out = (float*)d_out;

  const int N = in_sizes[0] / DIM_IN;
  const int E = in_sizes[1] / 2;
  const long long* src = edge;       // edge_index[0]
  const long long* dst = edge + E;   // edge_index[1]

  // Workspace layout: inv_deg[N] | agg[N*128] | h0[N*128] | h1[N*128]
  float* invd = (float*)d_ws;
  float* agg = invd + N;
  float* h0 = agg + (long long)N * DIM_IN;
  float* h1 = h0 + (long long)N * DIM_IN;

  const int ZB = 2048;
  const int degB = (E + 255) / 256;
  const int invB = (N + 255) / 256;
  const int scatB = (int)(((long long)E * 32 + 255) / 256);

  const int iter128 = (N + 15) / 16;          // RT=1 for DOUT=128
  const int iter64 = (N + 31) / 32;           // RT=2 for DOUT=64
  const int gemmB128 = iter128 < 768 ? iter128 : 768;
  const int gemmB64 = iter64 < 768 ? iter64 : 768;

  // degrees -> inverse degrees
  zero_f32<<<ZB, 256, 0, stream>>>(invd, N);
  degree_kernel<<<degB, 256, 0, stream>>>(dst, invd, E);
  inv_deg_kernel<<<invB, 256, 0, stream>>>(invd, N);

  // Layer 0: features -> h0 (relu)
  zero_f32<<<ZB, 256, 0, stream>>>(agg, (long long)N * DIM_IN);
  scatter_kernel<<<scatB, 256, 0, stream>>>(features, src, dst, agg, E);
  sage_layer_kernel<128, true><<<gemmB128, 256, 0, stream>>>(
      features, agg, invd, Ws0, Wn0, b0, h0, N);

  // Layer 1: h0 -> h1 (relu)
  zero_f32<<<ZB, 256, 0, stream>>>(agg, (long long)N * DIM_IN);
  scatter_kernel<<<scatB, 256, 0, stream>>>(h0, src, dst, agg, E);
  sage_layer_kernel<128, true><<<gemmB128, 256, 0, stream>>>(
      h0, agg, invd, Ws1, Wn1, b1, h1, N);

  // Layer 2: h1 -> out (no activation, Dout=64)
  zero_f32<<<ZB, 256, 0, stream>>>(agg, (long long)N * DIM_IN);
  scatter_kernel<<<scatB, 256, 0, stream>>>(h1, src, dst, agg, E);
  sage_layer_kernel<64, false><<<gemmB64, 256, 0, stream>>>(
      h1, agg, invd, Ws2, Wn2, b2, out, N);
}